// GraphSAGENet_79594333929755
// MI455X (gfx1250) — compile-verified
//
#include <hip/hip_runtime.h>
#include <hip/hip_bf16.h>
#include <stdint.h>

// ---------------- problem constants (from reference) ----------------
#define N_NODES 100000
#define N_EDGES 1600000
#define IN_C    165
#define HID_C   256
#define OUT_C   2

typedef float v2f __attribute__((ext_vector_type(2)));
typedef float v8f __attribute__((ext_vector_type(8)));

// ---------------------------------------------------------------
// Kernel 1: per-edge scatter-add of x[src] into agg[dst]; one wave
// (32 lanes) per edge so the 165-channel gather is coalesced.
// Lane 0 also accumulates the destination degree.
// ---------------------------------------------------------------
__global__ __launch_bounds__(256)
void k_scatter_x_deg(const float* __restrict__ x,
                     const long long* __restrict__ ei,
                     float* __restrict__ agg,
                     float* __restrict__ deg) {
    long long gid = (long long)blockIdx.x * blockDim.x + threadIdx.x;
    int e    = (int)(gid >> 5);
    int lane = (int)(gid & 31);
    if (e >= N_EDGES) return;
    int s = (int)ei[e];
    int d = (int)ei[(long long)N_EDGES + e];
    if (lane == 0) atomicAdd(&deg[d], 1.0f);
    const float* xs = x   + (size_t)s * IN_C;
    float*       ad = agg + (size_t)d * IN_C;
    for (int c = lane; c < IN_C; c += 32)
        atomicAdd(&ad[c], xs[c]);
}

// ---------------------------------------------------------------
// Kernel 2: in-place mean: agg[n][c] /= max(deg[n], 1)
// ---------------------------------------------------------------
__global__ __launch_bounds__(256)
void k_mean(float* __restrict__ agg, const float* __restrict__ deg) {
    long long idx = (long long)blockIdx.x * blockDim.x + threadIdx.x;
    if (idx >= (long long)N_NODES * IN_C) return;
    int n = (int)(idx / IN_C);
    agg[idx] = agg[idx] / fmaxf(deg[n], 1.0f);
}

// ---------------------------------------------------------------
// Kernel 3: layer-1 fused dual GEMM + bias + ReLU using fp32 WMMA.
//   h = relu(mean @ W1_l + x @ W1_r + b1)     [100000 x 256]
// One wave computes one 16x16 output tile; block = 8 waves = 8 N-tiles.
// grid = (N_NODES/16, HID_C/(16*8)) = (6250, 2).
//
// Fragment layouts per CDNA5 ISA 7.12.2 (wave32):
//   A (16x4, M x K):  lanes 0-15 -> M=lane, K={0,1}; lanes 16-31 -> M=lane-16, K={2,3}
//   B (4x16, K x N):  mirrored: lane gives N, half-wave selects K pair
//   C/D (16x16):      VGPR r: lanes 0-15 -> M=r, N=lane; lanes 16-31 -> M=8+r, N=lane-16
//
// Two independent accumulator chains (acc0/acc1) alternate K-steps so
// the WMMA pipeline always has two chains in flight instead of one
// 84-deep serial RAW chain; they are summed once at the end.
// K handling is branchless: steps with kk <= 160 are fully in-range
// (lane touches at most kk+3 = 163 < 165); the final K=164 step uses a
// v_cndmask-style 0/1 mask (no EXEC change — WMMA needs EXEC all-ones).
// ---------------------------------------------------------------
__device__ inline void gemm_tile_f32_dual(const float* __restrict__ A,   // [N_NODES x IN_C]
                                          const float* __restrict__ B,   // [IN_C x HID_C]
                                          int row, int col, int kpair,
                                          v8f& acc0, v8f& acc1) {
    const float* Arow = A + (size_t)row * IN_C + kpair;      // +4 per step
    const float* Bcol = B + (size_t)kpair * HID_C + col;     // +4*HID_C per step

    // 20 iterations x 2 interleaved steps: K = 0..159
    for (int it = 0; it < 20; ++it) {
        v2f a0, b0, a1, b1;
        a0.x = Arow[0];
        a0.y = Arow[1];
        b0.x = Bcol[0];
        b0.y = Bcol[HID_C];
        a1.x = Arow[4];
        a1.y = Arow[5];
        b1.x = Bcol[4 * HID_C];
        b1.y = Bcol[5 * HID_C];
        acc0 = __builtin_amdgcn_wmma_f32_16x16x4_f32(
            false, a0, false, b0, (short)0, acc0, false, false);
        acc1 = __builtin_amdgcn_wmma_f32_16x16x4_f32(
            false, a1, false, b1, (short)0, acc1, false, false);
        Arow += 8;
        Bcol += 8 * HID_C;
    }
    // K = 160..163 -> acc0 (fully in-range: max index 163 < 165)
    {
        v2f a, b;
        a.x = Arow[0];
        a.y = Arow[1];
        b.x = Bcol[0];
        b.y = Bcol[HID_C];
        acc0 = __builtin_amdgcn_wmma_f32_16x16x4_f32(
            false, a, false, b, (short)0, acc0, false, false);
    }
    // tail: K = 164 (only valid for kpair==0, first component) -> acc1
    {
        float mask = (kpair == 0) ? 1.0f : 0.0f;   // branchless select
        v2f a, b;
        a.x = A[(size_t)row * IN_C + (IN_C - 1)] * mask;   // always in-range load
        a.y = 0.0f;
        b.x = B[(size_t)(IN_C - 1) * HID_C + col];          // in-range; a.x==0 kills it
        b.y = 0.0f;
        acc1 = __builtin_amdgcn_wmma_f32_16x16x4_f32(
            false, a, false, b, (short)0, acc1, false, false);
    }
}

__global__ __launch_bounds__(256)
void k_layer1_wmma(const float* __restrict__ mean,
                   const float* __restrict__ x,
                   const float* __restrict__ W1l,
                   const float* __restrict__ W1r,
                   const float* __restrict__ b1,
                   float* __restrict__ h) {
    int lane  = threadIdx.x & 31;
    int wv    = threadIdx.x >> 5;
    int m0    = blockIdx.x * 16;
    int n0    = (blockIdx.y * 8 + wv) * 16;
    int row   = m0 + (lane & 15);
    int col   = n0 + (lane & 15);
    int kpair = (lane >> 4) << 1;          // 0 for lanes 0-15, 2 for lanes 16-31

    v8f acc0 = {};
    v8f acc1 = {};
    gemm_tile_f32_dual(mean, W1l, row, col, kpair, acc0, acc1);   // mean @ W1_l
    gemm_tile_f32_dual(x,    W1r, row, col, kpair, acc0, acc1);   // + x @ W1_r
    v8f acc = acc0 + acc1;

    float bias  = b1[col];
    int   mbase = m0 + ((lane >> 4) << 3);   // lanes 16-31 hold rows m0+8..m0+15
#pragma unroll
    for (int r = 0; r < 8; ++r) {
        float v = acc[r] + bias;
        v = fmaxf(v, 0.0f);                  // ReLU
        h[(size_t)(mbase + r) * HID_C + col] = v;
    }
}

// ---------------------------------------------------------------
// Kernel 4: layer-2 projections (project BEFORE aggregating — linearity
// of segment_sum — so edges only move 2 floats instead of 256):
//   t = h @ W2_l   [N x 2],   r = h @ W2_r   [N x 2]
// One wave per node; each lane owns 8 consecutive channels and loads
// everything as float4 (global_load_b128); wave32 shuffle reduction.
// ---------------------------------------------------------------
__global__ __launch_bounds__(256)
void k_proj2(const float* __restrict__ h,
             const float* __restrict__ W2l,
             const float* __restrict__ W2r,
             float* __restrict__ t,
             float* __restrict__ r) {
    long long gid = (long long)blockIdx.x * blockDim.x + threadIdx.x;
    int n    = (int)(gid >> 5);
    int lane = (int)(gid & 31);
    if (n >= N_NODES) return;

    // lane owns channels c = lane*8 .. lane*8+7
    const float4* h4  = (const float4*)(h + (size_t)n * HID_C) + lane * 2;   // 2 x float4
    const float4* wl4 = (const float4*)W2l + lane * 4;   // flat [lane*16 .. lane*16+15]
    const float4* wr4 = (const float4*)W2r + lane * 4;

    float4 hv0 = h4[0];                 // c = lane*8 + 0..3
    float4 hv1 = h4[1];                 // c = lane*8 + 4..7
    float hv[8] = {hv0.x, hv0.y, hv0.z, hv0.w, hv1.x, hv1.y, hv1.z, hv1.w};

    float tl0 = 0.f, tl1 = 0.f, rr0 = 0.f, rr1 = 0.f;
#pragma unroll
    for (int q = 0; q < 4; ++q) {       // each float4 covers 2 channels (interleaved ch0/ch1)
        float4 wl = wl4[q];
        float4 wr = wr4[q];
        float h0 = hv[2 * q + 0];
        float h1 = hv[2 * q + 1];
        tl0 += h0 * wl.x + h1 * wl.z;
        tl1 += h0 * wl.y + h1 * wl.w;
        rr0 += h0 * wr.x + h1 * wr.z;
        rr1 += h0 * wr.y + h1 * wr.w;
    }
#pragma unroll
    for (int off = 16; off; off >>= 1) {    // wave32 reduction
        tl0 += __shfl_xor(tl0, off, 32);
        tl1 += __shfl_xor(tl1, off, 32);
        rr0 += __shfl_xor(rr0, off, 32);
        rr1 += __shfl_xor(rr1, off, 32);
    }
    if (lane == 0) {
        t[n * 2 + 0] = tl0;  t[n * 2 + 1] = tl1;
        r[n * 2 + 0] = rr0;  r[n * 2 + 1] = rr1;
    }
}

// ---------------------------------------------------------------
// Kernel 5: scatter the 2-float projected messages over edges.
// ---------------------------------------------------------------
__global__ __launch_bounds__(256)
void k_scatter_t(const long long* __restrict__ ei,
                 const float* __restrict__ t,
                 float* __restrict__ aggt) {
    int e = blockIdx.x * blockDim.x + threadIdx.x;
    if (e >= N_EDGES) return;
    int s = (int)ei[e];
    int d = (int)ei[(long long)N_EDGES + e];
    atomicAdd(&aggt[d * 2 + 0], t[s * 2 + 0]);
    atomicAdd(&aggt[d * 2 + 1], t[s * 2 + 1]);
}

// ---------------------------------------------------------------
// Kernel 6: out = aggt/max(deg,1) + b2 + r
// ---------------------------------------------------------------
__global__ __launch_bounds__(256)
void k_final(const float* __restrict__ aggt,
             const float* __restrict__ deg,
             const float* __restrict__ b2,
             const float* __restrict__ r,
             float* __restrict__ out) {
    int n = blockIdx.x * blockDim.x + threadIdx.x;
    if (n >= N_NODES) return;
    float inv = 1.0f / fmaxf(deg[n], 1.0f);
    out[n * 2 + 0] = aggt[n * 2 + 0] * inv + b2[0] + r[n * 2 + 0];
    out[n * 2 + 1] = aggt[n * 2 + 1] * inv + b2[1] + r[n * 2 + 1];
}

// ---------------------------------------------------------------
static inline size_t align256(size_t v) { return (v + 255) & ~(size_t)255; }

extern "C" void kernel_launch(void* const* d_in, const int* in_sizes, int n_in,
                              void* d_out, int out_size, void* d_ws, size_t ws_size,
                              hipStream_t stream) {
    // inputs in setup_inputs() dict order
    const float*     x   = (const float*)d_in[0];
    const long long* ei  = (const long long*)d_in[1];   // int64 edge_index [2, E]
    const float*     W1l = (const float*)d_in[2];
    const float*     b1  = (const float*)d_in[3];
    const float*     W1r = (const float*)d_in[4];
    const float*     W2l = (const float*)d_in[5];
    const float*     b2  = (const float*)d_in[6];
    const float*     W2r = (const float*)d_in[7];
    float* out = (float*)d_out;

    // workspace carve-up
    char* ws = (char*)d_ws;
    size_t off = 0;
    float* deg  = (float*)(ws + off); off = align256(off + (size_t)N_NODES * sizeof(float));
    float* agg  = (float*)(ws + off); off = align256(off + (size_t)N_NODES * IN_C  * sizeof(float));
    float* h    = (float*)(ws + off); off = align256(off + (size_t)N_NODES * HID_C * sizeof(float));
    float* t    = (float*)(ws + off); off = align256(off + (size_t)N_NODES * OUT_C * sizeof(float));
    float* rbuf = (float*)(ws + off); off = align256(off + (size_t)N_NODES * OUT_C * sizeof(float));
    float* aggt = (float*)(ws + off); off = align256(off + (size_t)N_NODES * OUT_C * sizeof(float));
    (void)ws_size; (void)in_sizes; (void)n_in; (void)out_size;

    // zero accumulators (graph-capture-safe memset nodes)
    hipMemsetAsync(deg,  0, (size_t)N_NODES * sizeof(float), stream);
    hipMemsetAsync(agg,  0, (size_t)N_NODES * IN_C * sizeof(float), stream);
    hipMemsetAsync(aggt, 0, (size_t)N_NODES * OUT_C * sizeof(float), stream);

    // 1) edge scatter of x + degree  (wave per edge)
    {
        long long threads = (long long)N_EDGES * 32;
        int blocks = (int)((threads + 255) / 256);
        k_scatter_x_deg<<<blocks, 256, 0, stream>>>(x, ei, agg, deg);
    }
    // 2) mean (in place)
    {
        long long elems = (long long)N_NODES * IN_C;
        int blocks = (int)((elems + 255) / 256);
        k_mean<<<blocks, 256, 0, stream>>>(agg, deg);
    }
    // 3) layer-1 fused dual WMMA GEMM + bias + ReLU
    {
        dim3 grid(N_NODES / 16, HID_C / (16 * 8));   // (6250, 2)
        k_layer1_wmma<<<grid, 256, 0, stream>>>(agg, x, W1l, W1r, b1, h);
    }
    // 4) layer-2 projections (wave per node)
    {
        long long threads = (long long)N_NODES * 32;
        int blocks = (int)((threads + 255) / 256);
        k_proj2<<<blocks, 256, 0, stream>>>(h, W2l, W2r, t, rbuf);
    }
    // 5) scatter projected messages (thread per edge)
    {
        int blocks = (N_EDGES + 255) / 256;
        k_scatter_t<<<blocks, 256, 0, stream>>>(ei, t, aggt);
    }
    // 6) final combine
    {
        int blocks = (N_NODES + 255) / 256;
        k_final<<<blocks, 256, 0, stream>>>(aggt, deg, b2, rbuf, out);
    }
}